// WeightedAverageWirelength_24206435681020
// MI455X (gfx1250) — compile-verified
//
#include <hip/hip_runtime.h>
#include <hip/hip_bf16.h>

#define N_PINS 10000000
#define N_NETS 2000000

typedef __attribute__((ext_vector_type(2))) float v2f;
typedef __attribute__((ext_vector_type(8))) float v8f;

// Monotonic float<->uint mapping: float order == unsigned order.
__device__ __forceinline__ unsigned fenc(float f) {
    unsigned u = __float_as_uint(f);
    return (u & 0x80000000u) ? ~u : (u | 0x80000000u);
}
__device__ __forceinline__ float fdec(unsigned e) {
    return __uint_as_float((e & 0x80000000u) ? (e ^ 0x80000000u) : ~e);
}

// ---------------- init per-net scratch (and optionally the output scalar) ----
// mm[i]   = {enc(max coord), enc(max -coord)}   (uint2, one b64 store)
// abcd[i] = {a, b, c, d} segment sums            (float4, one b128 store)
__global__ __launch_bounds__(256) void wl_init(uint2* __restrict__ mm,
                                               float4* __restrict__ abcd,
                                               float* __restrict__ out, int zero_out) {
    int i = blockIdx.x * blockDim.x + threadIdx.x;
    if (i < N_NETS) {
        mm[i] = make_uint2(0u, 0u);   // 0 is below enc() of any value -> "empty net" flag
        abcd[i] = make_float4(0.f, 0.f, 0.f, 0.f);
    }
    if (zero_out && i == 0) *out = 0.f;
}

// ---------------- pass 1: per-net max(coord), max(-coord) -------------------
__global__ __launch_bounds__(256) void wl_minmax(const float* __restrict__ coord,
                                                 const int* __restrict__ p2n,
                                                 uint2* __restrict__ mm) {
    const int nvec = N_PINS / 4;
    const int stride = gridDim.x * blockDim.x;
    const float4* __restrict__ c4p = (const float4*)coord;
    const int4* __restrict__ n4p = (const int4*)p2n;
    for (int v = blockIdx.x * blockDim.x + threadIdx.x; v < nvec; v += stride) {
        int pf = v + stride;                       // prefetch next grid-stride chunk
        if (pf < nvec) {
            __builtin_prefetch(&c4p[pf], 0, 3);    // -> global_prefetch_b8
            __builtin_prefetch(&n4p[pf], 0, 3);
        }
        float4 c4 = c4p[v];
        int4 n4 = n4p[v];
        int ns[4] = {n4.x, n4.y, n4.z, n4.w};
        float cs[4] = {c4.x, c4.y, c4.z, c4.w};
#pragma unroll
        for (int k = 0; k < 4; ++k) {
            // both atomics land in the same cacheline of mm[net]
            atomicMax(&mm[ns[k]].x, fenc(cs[k]));
            atomicMax(&mm[ns[k]].y, fenc(-cs[k]));
        }
    }
}

// ---------------- pass 2: exp terms + 4 segment sums ------------------------
__global__ __launch_bounds__(256) void wl_sums(const float* __restrict__ coord,
                                               const int* __restrict__ p2n,
                                               const float* __restrict__ gamma,
                                               const uint2* __restrict__ mm,
                                               float4* __restrict__ abcd) {
    const float invg = 1.0f / *gamma;
    const int nvec = N_PINS / 4;
    const int stride = gridDim.x * blockDim.x;
    const float4* __restrict__ c4p = (const float4*)coord;
    const int4* __restrict__ n4p = (const int4*)p2n;
    for (int v = blockIdx.x * blockDim.x + threadIdx.x; v < nvec; v += stride) {
        int pf = v + stride;
        if (pf < nvec) {
            __builtin_prefetch(&c4p[pf], 0, 3);
            __builtin_prefetch(&n4p[pf], 0, 3);
        }
        float4 c4 = c4p[v];
        int4 n4 = n4p[v];
        int ns[4] = {n4.x, n4.y, n4.z, n4.w};
        float cs[4] = {c4.x, c4.y, c4.z, c4.w};
#pragma unroll
        for (int k = 0; k < 4; ++k) {
            int net = ns[k];
            float cc = cs[k];
            uint2 e = mm[net];                       // single b64 gather
            float cmax = fdec(e.x);
            float cmin = -fdec(e.y);
            float ep = __expf((cc - cmax) * invg);   // arg <= 0
            float en = __expf((cmin - cc) * invg);   // arg <= 0
            // all four atomics hit one 16B span of one cacheline
            atomicAdd(&abcd[net].x, ep);
            atomicAdd(&abcd[net].y, cc * ep);
            atomicAdd(&abcd[net].z, en);
            atomicAdd(&abcd[net].w, cc * en);
        }
    }
}

// ---------------- pass 3: per-net wl, WMMA-assisted block reduce ------------
__global__ __launch_bounds__(256) void wl_finalize(const uint2* __restrict__ mm,
                                                   const float4* __restrict__ abcd,
                                                   const unsigned char* __restrict__ net_mask,
                                                   float* __restrict__ out) {
    // NOTE: no early returns anywhere -> EXEC stays all-1s for the WMMA.
    int i = blockIdx.x * blockDim.x + threadIdx.x;
    float wl = 0.f;
    if (i < N_NETS) {
        unsigned me = mm[i].x;
        if (me != 0u && net_mask[i]) {
            float4 s4 = abcd[i];                     // single b128 load
            wl = s4.y / s4.x - s4.w / s4.z;
        }
    }

    // Wave32 reduction via V_WMMA_F32_16X16X4_F32 with B = ones:
    //   A[m][k]: lane m (0..15) holds K=0,1 in its 2 VGPRs, lane m+16 holds K=2,3.
    //   Partial in A-vgpr0, 0 in A-vgpr1 -> rowsum(m) = wl[lane m] + wl[lane m+16],
    //   and D[m][n] = rowsum(m) for every n.
    v2f A; A.x = wl; A.y = 0.0f;
    v2f B; B.x = 1.0f; B.y = 1.0f;
    v8f C = {};
    v8f D = __builtin_amdgcn_wmma_f32_16x16x4_f32(false, A, false, B, (short)0, C,
                                                  false, false);
    float s = D[0] + D[1] + D[2] + D[3] + D[4] + D[5] + D[6] + D[7];
    s += __shfl_xor(s, 16, 32);   // lanes 0..15 hold rows 0..7, lanes 16..31 rows 8..15

    __shared__ float wsum[8];     // 256 threads = 8 wave32
    int wave = threadIdx.x >> 5;
    int lane = threadIdx.x & 31;
    if (lane == 0) wsum[wave] = s;
    __syncthreads();
    if (threadIdx.x == 0) {
        float t = 0.f;
#pragma unroll
        for (int w = 0; w < 8; ++w) t += wsum[w];
        atomicAdd(out, t);        // global_atomic_add_f32
    }
}

extern "C" void kernel_launch(void* const* d_in, const int* in_sizes, int n_in,
                              void* d_out, int out_size, void* d_ws, size_t ws_size,
                              hipStream_t stream) {
    const float* pos = (const float*)d_in[0];
    const int* p2n = (const int*)d_in[1];
    const unsigned char* net_mask = (const unsigned char*)d_in[2];
    // d_in[3] = pin_mask (unused by the reference)
    const float* gamma = (const float*)d_in[4];
    float* out = (float*)d_out;

    // Workspace: 16 MB uint2 + 32 MB float4 = 48 MB, reused for x and y rounds.
    uint2* mm = (uint2*)d_ws;
    float4* abcd = (float4*)(mm + N_NETS);

    dim3 blk(256);
    dim3 grid_nets((N_NETS + 255) / 256);
    dim3 grid_pins(2048);   // grid-stride over 2.5M float4 elements

    for (int dimi = 0; dimi < 2; ++dimi) {
        const float* coord = pos + (size_t)dimi * N_PINS;
        wl_init<<<grid_nets, blk, 0, stream>>>(mm, abcd, out, dimi == 0);
        wl_minmax<<<grid_pins, blk, 0, stream>>>(coord, p2n, mm);
        wl_sums<<<grid_pins, blk, 0, stream>>>(coord, p2n, gamma, mm, abcd);
        wl_finalize<<<grid_nets, blk, 0, stream>>>(mm, abcd, net_mask, out);
    }
}